// PLEEncoder_23227183137574
// MI455X (gfx1250) — compile-verified
//
#include <hip/hip_runtime.h>

// PLE encoder: samples (8,64,2048) f32, bin_edges (64,65) f32 -> out (8, 64*64, 2048) f32.
// Store-bandwidth-bound (256 MB written once). No matmul structure -> no WMMA.
// CDNA5 paths used: TDM tensor_load_to_lds + s_wait_tensorcnt, non-temporal
// global_store_b128 streaming stores, global_prefetch, wave32 coalescing.

#define B_  8
#define C_  64
#define L_  2048
#define NB_ 64
#define NE_ 65

typedef __attribute__((ext_vector_type(4))) float        f32x4;
typedef __attribute__((ext_vector_type(4))) unsigned int u32x4;
typedef __attribute__((ext_vector_type(8))) int          i32x8;
typedef __attribute__((ext_vector_type(4))) int          i32x4;

__global__ __launch_bounds__(256) void ple_encode_kernel(
    const float* __restrict__ samples,
    const float* __restrict__ edges,
    float* __restrict__ out)
{
  __shared__ float s_edge[NE_];   // this channel's 65 bin edges
  __shared__ float s_invw[NB_];   // 1 / width per bin

  const int c  = blockIdx.y;
  const int b  = blockIdx.z;
  const int l0 = (blockIdx.x * 256 + threadIdx.x) * 4;   // 4 consecutive l per lane

  const float* sptr = samples + ((size_t)(b * C_ + c) * L_ + l0);
  // Overlap the streaming sample fetch with the edge-table DMA + barrier.
  __builtin_prefetch(sptr, 0, 0);   // -> global_prefetch_b8

#if defined(__gfx1250__) && __has_builtin(__builtin_amdgcn_tensor_load_to_lds)
  // One TDM descriptor per block: DMA 65 contiguous f32 (this channel's edges)
  // into LDS. Issue from wave 0 only; split-wait on TENSORcnt before the
  // workgroup barrier publishes the data to all 8 waves.
  if (threadIdx.x < 32) {
    const unsigned long long ga =
        (unsigned long long)(const void*)(edges + (size_t)c * NE_);
    const unsigned int lds_off =
        (unsigned int)(unsigned long long)(const void*)&s_edge[0];

    u32x4 g0;
    g0[0] = 1u;                                            // count=1 (valid), user mode
    g0[1] = lds_off;                                       // lds_addr [63:32]
    g0[2] = (unsigned int)(ga & 0xffffffffu);              // global_addr[31:0]
    g0[3] = (unsigned int)((ga >> 32) & 0x1ffffffu)        // global_addr[56:32]
          | (2u << 30);                                    // type=2 ("image")

    i32x8 g1;
    g1[0] = (int)(2u << 16);                // workgroup_mask=0, data_size=2 (4B)
    g1[1] = (int)((unsigned)NE_ << 16);     // tensor_dim0=65 (low 16 bits @ [63:48])
    g1[2] = (int)(1u << 16);                // tensor_dim0 hi=0, tensor_dim1=1 @ [95:80]
    g1[3] = (int)((unsigned)NE_ << 16);     // tile_dim0=65 @ [127:112]
    g1[4] = 1;                              // tile_dim1=1, tile_dim2=0
    g1[5] = NE_;                            // tensor_dim0_stride=65
    g1[6] = 0;                              // stride hi / dim1_stride lo
    g1[7] = 0;                              // dim1_stride hi
    i32x4 gz4 = {0, 0, 0, 0};               // <=2D tensor: groups 2/3 unused
    i32x8 gz8 = {0, 0, 0, 0, 0, 0, 0, 0};   // extra group (clang-23 6-arg form)

    __builtin_amdgcn_tensor_load_to_lds(g0, g1, gz4, gz4, gz8, 0);
    __builtin_amdgcn_s_wait_tensorcnt(0);   // -> s_wait_tensorcnt 0
  }
#else
  if (threadIdx.x < NE_) s_edge[threadIdx.x] = edges[(size_t)c * NE_ + threadIdx.x];
#endif
  __syncthreads();

  // Precompute reciprocal bin widths once per block (no divides on hot path).
  if (threadIdx.x < NB_)
    s_invw[threadIdx.x] = 1.0f / (s_edge[threadIdx.x + 1] - s_edge[threadIdx.x]);
  __syncthreads();

  const f32x4 x4 = *reinterpret_cast<const f32x4*>(sptr);

  int   bin[4];
  float val[4];
#pragma unroll
  for (int s = 0; s < 4; ++s) {
    const float x = x4[s];
    // searchsorted(inner, x, 'right') over the 63 inner edges (edge[1..63]):
    // branch-free binary search; range size is exactly 64 = 2^6, so 6 exact
    // halvings, never probing past edge[63]. Result bin in [0, 63].
    int lo = 0, hi = NB_ - 1;
#pragma unroll
    for (int it = 0; it < 6; ++it) {
      const int  mid = (lo + hi) >> 1;
      const bool ge  = (s_edge[mid + 1] <= x);   // inner[mid] <= x
      lo = ge ? (mid + 1) : lo;
      hi = ge ? hi        : mid;
    }
    bin[s] = lo;
    val[s] = (x - s_edge[lo]) * s_invw[lo];
  }

  // out[b, c*NB + j, l]; per-j store: lane t covers l0..l0+3 -> 512B/wave,
  // j*L_*4 <= 516096 fits the signed 24-bit immediate offset.
  float* optr = out + ((size_t)(b * C_ + c) * NB_) * L_ + l0;
#pragma unroll
  for (int j = 0; j < NB_; ++j) {
    f32x4 o;
#pragma unroll
    for (int s = 0; s < 4; ++s)
      o[s] = (j < bin[s]) ? 1.0f : ((j == bin[s]) ? val[s] : 0.0f);
    // 256 MB written once, never re-read by this kernel; output > 192 MB L2:
    // stream with non-temporal hint.
    __builtin_nontemporal_store(o, reinterpret_cast<f32x4*>(optr + (size_t)j * L_));
  }
}

extern "C" void kernel_launch(void* const* d_in, const int* in_sizes, int n_in,
                              void* d_out, int out_size, void* d_ws, size_t ws_size,
                              hipStream_t stream) {
  const float* samples = (const float*)d_in[0];   // (8, 64, 2048) f32
  const float* edges   = (const float*)d_in[1];   // (64, 65) f32
  float*       out     = (float*)d_out;           // (8, 4096, 2048) f32

  dim3 grid(L_ / (256 * 4), C_, B_);              // (2, 64, 8) blocks
  ple_encode_kernel<<<grid, dim3(256), 0, stream>>>(samples, edges, out);
}